// PixelArt_32856499814716
// MI455X (gfx1250) — compile-verified
//
#include <hip/hip_runtime.h>
#include <stdint.h>

// out[i,j,:] = palette[argmax_k img_cls[i,j,k], :]   (forward value of the
// straight-through estimator: st == hard one-hot, softmax is monotone).
// Pure streaming argmax-gather: 1 GiB read @ 23.3 TB/s => ~47 us floor.

typedef float v4f __attribute__((ext_vector_type(4)));

#define KCLS   256          // classes per pixel (contiguous fp32)
#define WAVESZ 32           // CDNA5 is wave32
#define NWAVES 8            // 256-thread block = 8 waves
#define PPW    4            // pixels per wave
#define PPB    (NWAVES * PPW)

__global__ __launch_bounds__(256) void pixelart_argmax_gather(
    const float* __restrict__ img,   // [npix, 256]
    const float* __restrict__ pal,   // [256, 3]
    float* __restrict__ out,         // [npix, 3]
    int npix)
{
    __shared__ float s_pal[KCLS * 3];            // 3 KiB palette tile in LDS

    const int tid  = threadIdx.x;
    const int lane = tid & (WAVESZ - 1);
    const int wave = tid >> 5;

    // ---- CDNA5 async global->LDS copy of the palette (ASYNCcnt path) ----
    // 768 floats = 192 lanes x b128. LDS byte offset = low 32 bits of the
    // generic pointer (ISA 10.2).
    if (tid < (KCLS * 3) / 4) {
        uint32_t lds_addr = (uint32_t)(uintptr_t)(&s_pal[tid * 4]);
        const v4f* gsrc   = (const v4f*)pal + tid;
        asm volatile("global_load_async_to_lds_b128 %0, %1, off"
                     :: "v"(lds_addr), "v"(gsrc)
                     : "memory");
    }
    asm volatile("s_wait_asynccnt 0x0" ::: "memory");
    __syncthreads();

    const int base = blockIdx.x * PPB;

    for (int t = 0; t < PPW; ++t) {
        const int p = base + t * NWAVES + wave;   // wave-uniform pixel id
        if (p >= npix) break;

        const v4f* row = (const v4f*)(img + (size_t)p * KCLS);

        // Prefetch this wave's next row (emits global_prefetch_b8).
        if (t + 1 < PPW) {
            __builtin_prefetch(
                (const void*)((const float*)row + (size_t)NWAVES * KCLS + lane * 4),
                0, 0);
        }

        // Non-temporal streaming loads: 2 x b128 per lane = full 1 KiB row
        // per wave; single-pass data must not displace the 192 MB L2.
        v4f a0 = __builtin_nontemporal_load(row + lane);           // k = lane*4 ..
        v4f a1 = __builtin_nontemporal_load(row + WAVESZ + lane);  // k = 128 + lane*4 ..

        // Local argmax in increasing-k order (strict '>' keeps the FIRST
        // maximum, matching jnp.argmax tie-breaking).
        float bv = a0.x; int bi = lane * 4;
        if (a0.y > bv) { bv = a0.y; bi = lane * 4 + 1; }
        if (a0.z > bv) { bv = a0.z; bi = lane * 4 + 2; }
        if (a0.w > bv) { bv = a0.w; bi = lane * 4 + 3; }
        const int k2 = 128 + lane * 4;
        if (a1.x > bv) { bv = a1.x; bi = k2;     }
        if (a1.y > bv) { bv = a1.y; bi = k2 + 1; }
        if (a1.z > bv) { bv = a1.z; bi = k2 + 2; }
        if (a1.w > bv) { bv = a1.w; bi = k2 + 3; }

        // Wave32 butterfly reduction; explicit smaller-index tie-break.
        #pragma unroll
        for (int m = 16; m >= 1; m >>= 1) {
            float ov = __shfl_xor(bv, m, WAVESZ);
            int   oi = __shfl_xor(bi, m, WAVESZ);
            if (ov > bv || (ov == bv && oi < bi)) { bv = ov; bi = oi; }
        }

        // Gather 3 palette floats from LDS; lanes 0..2 each write one.
        if (lane < 3) {
            out[(size_t)p * 3 + lane] = s_pal[bi * 3 + lane];
        }
    }
}

extern "C" void kernel_launch(void* const* d_in, const int* in_sizes, int n_in,
                              void* d_out, int out_size, void* d_ws, size_t ws_size,
                              hipStream_t stream) {
    const float* img = (const float*)d_in[0];   // [H*W, 256] fp32
    const float* pal = (const float*)d_in[1];   // [256, 3]  fp32
    float* out = (float*)d_out;                 // [H*W, 3]  fp32

    const int npix   = in_sizes[0] / KCLS;
    const int blocks = (npix + PPB - 1) / PPB;

    pixelart_argmax_gather<<<blocks, 256, 0, stream>>>(img, pal, out, npix);
}